// MincutPoolingLayer_36515811951303
// MI455X (gfx1250) — compile-verified
//
#include <hip/hip_runtime.h>
#include <math.h>

typedef __attribute__((ext_vector_type(2))) float    v2f;
typedef __attribute__((ext_vector_type(8))) float    v8f;
typedef __attribute__((ext_vector_type(4))) unsigned v4u;
typedef __attribute__((ext_vector_type(4))) int      v4i;
typedef __attribute__((ext_vector_type(8))) int      v8i;

#define B_GRAPHS    64
#define NODES_PER_G 1600
#define N_NODES     (B_GRAPHS * NODES_PER_G)   // 102400
#define F_DIM       128
#define K_CL        16
#define CHUNK       200                         // nodes per chunk
#define NCHUNK      8                           // chunks per graph

// ---- workspace layout (float offsets). Zeroed prefix: [deg][M][scal][oxa][adj][cc] ----
#define OFF_DEG   0
#define OFF_M     (N_NODES)
#define OFF_SCAL  (OFF_M + N_NODES * K_CL)          // [0]=num [1]=den [2]=ortho
#define OFF_OXA   (OFF_SCAL + 4)                    // out_x accumulator B*K*F
#define OFF_ADJ   (OFF_OXA + B_GRAPHS * K_CL * F_DIM)
#define OFF_CC    (OFF_ADJ + B_GRAPHS * K_CL * K_CL)
#define ZERO_CNT  (OFF_CC + B_GRAPHS * K_CL * K_CL)
#define OFF_S     (ZERO_CNT)                        // divisible by 4 -> 16B aligned

// ---- output layout (floats) ----
#define OUT_X_OFF    0
#define OUT_ADJ_OFF  (B_GRAPHS * K_CL * F_DIM)                 // 131072
#define OUT_LOSS_OFF (OUT_ADJ_OFF + B_GRAPHS * K_CL * K_CL)    // 147456

__device__ __forceinline__ v8f wmma_f32(v2f a, v2f b, v8f c) {
  return __builtin_amdgcn_wmma_f32_16x16x4_f32(false, a, false, b, (short)0, c, false, false);
}

// in-wave XOR shuffle via ds_swizzle (group-of-32: and=0x1f, or=0, xor=MASK)
template <int MASK>
__device__ __forceinline__ float swz_xor(float v) {
  return __int_as_float(
      __builtin_amdgcn_ds_swizzle(__float_as_int(v), (MASK << 10) | 0x1f));
}

__device__ __forceinline__ float wave_reduce_sum(float v) {
  v += swz_xor<16>(v); v += swz_xor<8>(v); v += swz_xor<4>(v);
  v += swz_xor<2>(v);  v += swz_xor<1>(v);
  return v;
}

__device__ __forceinline__ float block_reduce_sum(float v, float* shm) {
  const int lane = threadIdx.x & 31;
  const int w    = threadIdx.x >> 5;
  v = wave_reduce_sum(v);
  __syncthreads();
  if (lane == 0) shm[w] = v;
  __syncthreads();
  float r = (threadIdx.x < 8) ? shm[threadIdx.x] : 0.f;
  if (w == 0) {
    r += swz_xor<4>(r); r += swz_xor<2>(r); r += swz_xor<1>(r);
    if (lane == 0) shm[0] = r;
  }
  __syncthreads();
  return shm[0];
}

// ---- TDM: 1-D DMA of nelem f32 from global into LDS (issue from ONE wave) ----
__device__ __forceinline__ void tdm_load_1d(unsigned lds_off, const void* gptr,
                                            unsigned nelem) {
  unsigned long long ga = (unsigned long long)(uintptr_t)gptr;
  v4u g0;
  g0[0] = 1u;                                        // count=1 (valid), user mode
  g0[1] = lds_off;                                   // lds_addr
  g0[2] = (unsigned)(ga & 0xffffffffu);              // global_addr[31:0]
  g0[3] = (unsigned)((ga >> 32) & 0x01ffffffu)       // global_addr[56:32]
        | (2u << 30);                                // type=2 ("image")
  v8i g1;
  g1[0] = (int)(2u << 16);                           // data_size=4B; no multicast/pad
  g1[1] = (int)((nelem & 0xffffu) << 16);            // tensor_dim0[15:0]
  g1[2] = (int)((nelem >> 16) | (1u << 16));         // tensor_dim0[31:16], tensor_dim1=1
  g1[3] = (int)((nelem & 0xffffu) << 16);            // tile_dim0 = nelem
  g1[4] = 0;                                         // tile_dim1=0, tile_dim2=0
  g1[5] = (int)nelem;                                // tensor_dim0_stride[31:0]
  g1[6] = 0;
  g1[7] = 0;
  v4i z4 = {0, 0, 0, 0};
#if defined(__clang_major__) && (__clang_major__ >= 23)
  v8i z8 = {0, 0, 0, 0, 0, 0, 0, 0};
  __builtin_amdgcn_tensor_load_to_lds(g0, g1, z4, z4, z8, 0);
#else
  __builtin_amdgcn_tensor_load_to_lds(g0, g1, z4, z4, 0);
#endif
}

// ---------------- zero-init scratch ----------------
__global__ void k_zero(float* __restrict__ p, int n) {
  for (int i = blockIdx.x * blockDim.x + threadIdx.x; i < n; i += gridDim.x * blockDim.x)
    p[i] = 0.f;
}

// ---------------- s = softmax(softmax(x @ W + b)) via f32 WMMA ----------------
// 8 waves / block, one 16-node tile per wave
__global__ void k_xw_softmax(const float* __restrict__ x, const float* __restrict__ W,
                             const float* __restrict__ bvec, float* __restrict__ s) {
  const int tile = blockIdx.x * 8 + (threadIdx.x >> 5);
  const int lane = threadIdx.x & 31;
  const int half = lane >> 4;
  const int n    = lane & 15;
  const float2* xrow = (const float2*)(x + (size_t)(tile * 16 + n) * F_DIM);
  v8f c = {};
  #pragma unroll 4
  for (int k0 = 0; k0 < F_DIM; k0 += 4) {
    float2 av = xrow[(k0 >> 1) + half];              // A[m=n][k0+2h .. +1]
    v2f a;  a.x = av.x;  a.y = av.y;
    v2f b;
    b.x = W[(size_t)(k0 + 2 * half + 0) * K_CL + n]; // B[k][col=n]
    b.y = W[(size_t)(k0 + 2 * half + 1) * K_CL + n];
    c = wmma_f32(a, b, c);
  }
  const float bn = bvec[n];
  #pragma unroll
  for (int r = 0; r < 8; r++) {
    float p = c[r] + bn;                             // row m=r+8*half, col n
    float mx = p;
    mx = fmaxf(mx, swz_xor<8>(mx)); mx = fmaxf(mx, swz_xor<4>(mx));
    mx = fmaxf(mx, swz_xor<2>(mx)); mx = fmaxf(mx, swz_xor<1>(mx));
    float e = __expf(p - mx);
    float sm = e;
    sm += swz_xor<8>(sm); sm += swz_xor<4>(sm);
    sm += swz_xor<2>(sm); sm += swz_xor<1>(sm);
    float s1 = e / sm;
    float mx2 = s1;
    mx2 = fmaxf(mx2, swz_xor<8>(mx2)); mx2 = fmaxf(mx2, swz_xor<4>(mx2));
    mx2 = fmaxf(mx2, swz_xor<2>(mx2)); mx2 = fmaxf(mx2, swz_xor<1>(mx2));
    float e2 = __expf(s1 - mx2);
    float sm2 = e2;
    sm2 += swz_xor<8>(sm2); sm2 += swz_xor<4>(sm2);
    sm2 += swz_xor<2>(sm2); sm2 += swz_xor<1>(sm2);
    const int m = r + 8 * half;
    s[(size_t)(tile * 16 + m) * K_CL + n] = e2 / sm2;
  }
}

// ---------------- deg + M = A@S : per-edge gather/scatter (L2-resident) ----------------
__global__ void k_edges(const float* __restrict__ ew, const int* __restrict__ esrc,
                        const int* __restrict__ edst, const float* __restrict__ s,
                        float* __restrict__ deg, float* __restrict__ M, int E) {
  const int e = blockIdx.x * blockDim.x + threadIdx.x;
  if (e >= E) return;
  const float w = ew[e];
  const int src = esrc[e];
  const int dst = edst[e];
  atomicAdd(&deg[dst], w);
  const float4* sv = (const float4*)(s + (size_t)dst * K_CL);
  float* mrow = M + (size_t)src * K_CL;
  #pragma unroll
  for (int q = 0; q < 4; q++) {
    float4 t = sv[q];
    atomicAdd(&mrow[4 * q + 0], w * t.x);
    atomicAdd(&mrow[4 * q + 1], w * t.y);
    atomicAdd(&mrow[4 * q + 2], w * t.z);
    atomicAdd(&mrow[4 * q + 3], w * t.w);
  }
}

// ---------------- mincut denominator: sum deg[i] * ||s_i||^2 ----------------
__global__ void k_den(const float* __restrict__ s, const float* __restrict__ deg,
                      float* __restrict__ scal) {
  __shared__ float shm[8];
  const int i = blockIdx.x * blockDim.x + threadIdx.x;
  float v = 0.f;
  if (i < N_NODES) {
    const float4* sv = (const float4*)(s + (size_t)i * K_CL);
    float ss = 0.f;
    #pragma unroll
    for (int q = 0; q < 4; q++) {
      float4 t = sv[q];
      ss += t.x * t.x + t.y * t.y + t.z * t.z + t.w * t.w;
    }
    v = deg[i] * ss;
  }
  v = block_reduce_sum(v, shm);
  if (threadIdx.x == 0) atomicAdd(&scal[1], v);
}

// ---------------- out_x partials: S_chunk^T X_chunk, TDM-staged s, WMMA ----------------
// grid = 64 graphs * 8 chunks, 8 waves; wave ft owns feature cols [16ft,16ft+16)
__global__ void k_outx_acc(const float* __restrict__ s, const float* __restrict__ x,
                           float* __restrict__ oxa) {
  __shared__ float s_lds[CHUNK * K_CL];              // 12.8 KB
  const int b     = blockIdx.x >> 3;
  const int chunk = blockIdx.x & 7;
  const int ft    = threadIdx.x >> 5;
  const int lane  = threadIdx.x & 31;
  const int half  = lane >> 4;
  const int n     = lane & 15;
  const int base  = b * NODES_PER_G + chunk * CHUNK;

  if (ft == 0) {                                     // wave 0 drives the TDM
    tdm_load_1d((unsigned)(uintptr_t)s_lds, s + (size_t)base * K_CL, CHUNK * K_CL);
    __builtin_amdgcn_s_wait_tensorcnt(0);
  }
  __syncthreads();

  v8f c = {};
  #pragma unroll 2
  for (int k0 = 0; k0 < CHUNK; k0 += 4) {
    const int kl = k0 + 2 * half;
    v2f a, bm;
    a.x = s_lds[(kl + 0) * K_CL + n];                // A[m=n][k] = s[node_k][m]
    a.y = s_lds[(kl + 1) * K_CL + n];
    bm.x = x[(size_t)(base + kl + 0) * F_DIM + ft * 16 + n];
    bm.y = x[(size_t)(base + kl + 1) * F_DIM + ft * 16 + n];
    c = wmma_f32(a, bm, c);
  }
  #pragma unroll
  for (int r = 0; r < 8; r++) {
    const int m = r + 8 * half;
    atomicAdd(&oxa[(size_t)(b * K_CL + m) * F_DIM + ft * 16 + n], c[r]);
  }
}

// ---------------- SELU epilogue for out_x ----------------
__global__ void k_selu(const float* __restrict__ acc, float* __restrict__ out) {
  const int i = blockIdx.x * blockDim.x + threadIdx.x;
  if (i < B_GRAPHS * K_CL * F_DIM) {
    const float alpha = 1.6732632423543772f, scale = 1.0507009873554805f;
    float v = acc[i];
    out[i] = scale * (v > 0.f ? v : alpha * (__expf(v) - 1.f));
  }
}

// ---------------- out_adj / CC partials: S^T M and S^T S, TDM-staged s ----------------
// grid = 64 graphs * 8 chunks, 2 waves: wave0 -> adj, wave1 -> CC
__global__ void k_adj_cc(const float* __restrict__ s, const float* __restrict__ M,
                         float* __restrict__ adj, float* __restrict__ cc) {
  __shared__ float s_lds[CHUNK * K_CL];
  const int b     = blockIdx.x >> 3;
  const int chunk = blockIdx.x & 7;
  const int wave  = threadIdx.x >> 5;
  const int lane  = threadIdx.x & 31;
  const int half  = lane >> 4;
  const int n     = lane & 15;
  const int base  = b * NODES_PER_G + chunk * CHUNK;

  if (wave == 0) {
    tdm_load_1d((unsigned)(uintptr_t)s_lds, s + (size_t)base * K_CL, CHUNK * K_CL);
    __builtin_amdgcn_s_wait_tensorcnt(0);
  }
  __syncthreads();

  v8f c = {};
  if (wave == 0) {                                   // S^T M : B from global M
    #pragma unroll 2
    for (int k0 = 0; k0 < CHUNK; k0 += 4) {
      const int kl = k0 + 2 * half;
      v2f a, bm;
      a.x = s_lds[(kl + 0) * K_CL + n];
      a.y = s_lds[(kl + 1) * K_CL + n];
      bm.x = M[(size_t)(base + kl + 0) * K_CL + n];
      bm.y = M[(size_t)(base + kl + 1) * K_CL + n];
      c = wmma_f32(a, bm, c);
    }
    #pragma unroll
    for (int r = 0; r < 8; r++)
      atomicAdd(&adj[(size_t)b * 256 + (r + 8 * half) * K_CL + n], c[r]);
  } else {                                           // S^T S : both operands from LDS
    #pragma unroll 2
    for (int k0 = 0; k0 < CHUNK; k0 += 4) {
      const int kl = k0 + 2 * half;
      v2f a, bm;
      a.x = s_lds[(kl + 0) * K_CL + n];
      a.y = s_lds[(kl + 1) * K_CL + n];
      bm.x = a.x;
      bm.y = a.y;
      c = wmma_f32(a, bm, c);
    }
    #pragma unroll
    for (int r = 0; r < 8; r++)
      atomicAdd(&cc[(size_t)b * 256 + (r + 8 * half) * K_CL + n], c[r]);
  }
}

// ---------------- per-graph epilogue: trace, ortho, degree-normalize ----------------
__global__ void k_finalize(const float* __restrict__ adj, const float* __restrict__ cc,
                           float* __restrict__ adj_out, float* __restrict__ scal) {
  __shared__ float shm[8];
  __shared__ float dpool[K_CL];
  const int b = blockIdx.x;
  const int tid = threadIdx.x;                       // 256: one per (m, n)
  const int m = tid >> 4, n = tid & 15;
  const float a0 = adj[(size_t)b * 256 + tid];
  const float c0 = cc[(size_t)b * 256 + tid];

  float fro2 = block_reduce_sum(c0 * c0, shm);
  float ccn  = sqrtf(fro2);
  float diff = c0 / ccn - ((m == n) ? 0.25f : 0.f);  // 1/sqrt(K)=0.25
  float dn2  = block_reduce_sum(diff * diff, shm);
  float tr   = block_reduce_sum((m == n) ? a0 : 0.f, shm);
  if (tid == 0) {
    atomicAdd(&scal[0], tr);
    atomicAdd(&scal[2], sqrtf(dn2));
  }
  float am = (m == n) ? 0.f : a0;
  float rs = am;                                     // row sum over n (16-lane groups)
  rs += swz_xor<8>(rs); rs += swz_xor<4>(rs);
  rs += swz_xor<2>(rs); rs += swz_xor<1>(rs);
  if (n == 0) dpool[m] = sqrtf(rs) + 1e-12f;
  __syncthreads();
  adj_out[(size_t)b * 256 + tid] = am / (dpool[m] * dpool[n]);
}

// ---------------- scalar losses ----------------
__global__ void k_loss(const float* __restrict__ scal, float* __restrict__ out) {
  if (threadIdx.x == 0) {
    out[0] = -scal[0] / scal[1];                     // mincut_loss
    out[1] = scal[2] / (float)B_GRAPHS;              // ortho_loss
  }
}

extern "C" void kernel_launch(void* const* d_in, const int* in_sizes, int n_in,
                              void* d_out, int out_size, void* d_ws, size_t ws_size,
                              hipStream_t stream) {
  const float* x    = (const float*)d_in[0];
  const float* W    = (const float*)d_in[1];
  const float* bv   = (const float*)d_in[2];
  const float* ew   = (const float*)d_in[3];
  const int*   esrc = (const int*)d_in[4];
  const int*   edst = (const int*)d_in[5];
  // d_in[6] = batch: graphs are uniform size, node i -> graph i/NODES_PER_G
  const int E = in_sizes[3];

  float* ws  = (float*)d_ws;
  float* out = (float*)d_out;
  float* deg  = ws + OFF_DEG;
  float* M    = ws + OFF_M;
  float* scal = ws + OFF_SCAL;
  float* oxa  = ws + OFF_OXA;
  float* adj  = ws + OFF_ADJ;
  float* cc   = ws + OFF_CC;
  float* s    = ws + OFF_S;

  k_zero<<<8192, 256, 0, stream>>>(ws, ZERO_CNT);
  k_xw_softmax<<<(N_NODES / 16) / 8, 256, 0, stream>>>(x, W, bv, s);
  k_edges<<<(E + 255) / 256, 256, 0, stream>>>(ew, esrc, edst, s, deg, M, E);
  k_den<<<(N_NODES + 255) / 256, 256, 0, stream>>>(s, deg, scal);
  k_outx_acc<<<B_GRAPHS * NCHUNK, 256, 0, stream>>>(s, x, oxa);
  k_selu<<<(B_GRAPHS * K_CL * F_DIM + 255) / 256, 256, 0, stream>>>(oxa, out + OUT_X_OFF);
  k_adj_cc<<<B_GRAPHS * NCHUNK, 64, 0, stream>>>(s, M, adj, cc);
  k_finalize<<<B_GRAPHS, 256, 0, stream>>>(adj, cc, out + OUT_ADJ_OFF, scal);
  k_loss<<<1, 32, 0, stream>>>(scal, out + OUT_LOSS_OFF);
}